// SphericalHarmonicEdgeAngularAttrs_20693152432401
// MI455X (gfx1250) — compile-verified
//
#include <hip/hip_runtime.h>

// ---------------------------------------------------------------------------
// SH(l<=3) edge attributes on MI455X (gfx1250, wave32).
//
// out[e][0..15] = C(16x20) * monomials20(u_e),  u_e = normalize(pos[dst]-pos[src])
//
// The 16x20 coefficient contraction is done with chained
// V_WMMA_F32_16X16X4_F32 (5 WMMAs, K=20), two 16-edge tiles per wave.
// Memory: pos (6 MB) is L2-resident -> gathers cheap; 512 MB output stream
// uses non-temporal stores to avoid evicting it.
// ---------------------------------------------------------------------------

typedef __attribute__((ext_vector_type(2))) float v2f;
typedef __attribute__((ext_vector_type(4))) float v4f;
typedef __attribute__((ext_vector_type(8))) float v8f;

// Monomial column order (K index 0..19):
//  0:1  1:x  2:y  3:z  4:x2  5:y2  6:z2  7:xy  8:xz  9:yz
// 10:x3 11:y3 12:z3 13:x2y 14:x2z 15:xy2 16:y2z 17:xz2 18:yz2 19:xyz
__device__ __constant__ float SH_C[16][20] = {
  //        1          x          y          z         x2         y2         z2         xy         xz         yz         x3         y3         z3        x2y        x2z        xy2        y2z        xz2        yz2        xyz
  { 1.0000000f,        0,         0,         0,         0,         0,         0,         0,         0,         0,         0,         0,         0,         0,         0,         0,         0,         0,         0,         0 }, // sh0
  {        0, 1.7320508f,         0,         0,         0,         0,         0,         0,         0,         0,         0,         0,         0,         0,         0,         0,         0,         0,         0,         0 }, // sh1
  {        0,         0, 1.7320508f,         0,         0,         0,         0,         0,         0,         0,         0,         0,         0,         0,         0,         0,         0,         0,         0,         0 }, // sh2
  {        0,         0,         0, 1.7320508f,         0,         0,         0,         0,         0,         0,         0,         0,         0,         0,         0,         0,         0,         0,         0,         0 }, // sh3
  {        0,         0,         0,         0,         0,         0,         0,         0, 3.8729833f,         0,         0,         0,         0,         0,         0,         0,         0,         0,         0,         0 }, // sh4 = sqrt15*xz
  {        0,         0,         0,         0,         0,         0,         0, 3.8729833f,         0,         0,         0,         0,         0,         0,         0,         0,         0,         0,         0,         0 }, // sh5 = sqrt15*xy
  {        0,         0,         0,         0,-1.1180340f, 2.2360680f,-1.1180340f,         0,         0,         0,         0,         0,         0,         0,         0,         0,         0,         0,         0,         0 }, // sh6
  {        0,         0,         0,         0,         0,         0,         0,         0,         0, 3.8729833f,         0,         0,         0,         0,         0,         0,         0,         0,         0,         0 }, // sh7 = sqrt15*yz
  {        0,         0,         0,         0,-1.9364917f,         0, 1.9364917f,         0,         0,         0,         0,         0,         0,         0,         0,         0,         0,         0,         0,         0 }, // sh8
  {        0,         0,         0,         0,         0,         0,         0,         0,         0,         0,-2.0916501f,         0,         0,         0,         0,         0,         0, 6.2749502f,         0,         0 }, // sh9
  {        0,         0,         0,         0,         0,         0,         0,         0,         0,         0,         0,         0,         0,         0,         0,         0,         0,         0,         0,10.2469508f }, // sh10
  {        0,         0,         0,         0,         0,         0,         0,         0,         0,         0,-1.6201852f,         0,         0,         0,         0, 6.4807407f,         0,-1.6201852f,         0,         0 }, // sh11
  {        0,         0,         0,         0,         0,         0,         0,         0,         0,         0,         0, 2.6457513f,         0,-3.9686270f,         0,         0,         0,         0,-3.9686270f,         0 }, // sh12
  {        0,         0,         0,         0,         0,         0,         0,         0,         0,         0,         0,         0,-1.6201852f,         0,-1.6201852f,         0, 6.4807407f,         0,         0,         0 }, // sh13
  {        0,         0,         0,         0,         0,         0,         0,         0,         0,         0,         0,         0,         0,-5.1234754f,         0,         0,         0,         0, 5.1234754f,         0 }, // sh14
  {        0,         0,         0,         0,         0,         0,         0,         0,         0,         0,         0,         0, 2.0916501f,         0,-6.2749502f,         0,         0,         0,         0,         0 }, // sh15
};

__global__ __launch_bounds__(256)
void sh_l3_edge_kernel(const float* __restrict__ pos,
                       const int* __restrict__ edge_index,
                       float* __restrict__ out,
                       int E)
{
    const int lane = threadIdx.x & 31;
    const int wave = threadIdx.x >> 5;
    const long long tile = (long long)blockIdx.x * 8 + wave;
    const long long eBase = tile * 32;

    // ---- per-lane edge: gather + normalize (clamped so EXEC stays full) ----
    long long e = eBase + lane;
    long long ec = e < E ? e : (long long)E - 1;
    const int src = __builtin_nontemporal_load(edge_index + ec);        // streamed once
    const int dst = __builtin_nontemporal_load(edge_index + (long long)E + ec);

    const float* ps = pos + 3ll * src;   // pos table is 6 MB -> L2-resident
    const float* pd = pos + 3ll * dst;
    const float vx = pd[0] - ps[0];
    const float vy = pd[1] - ps[1];
    const float vz = pd[2] - ps[2];
    const float inv = __builtin_amdgcn_rsqf(vx * vx + vy * vy + vz * vz); // v_rsq_f32
    const float x = vx * inv, y = vy * inv, z = vz * inv;

    // ---- 20 monomials of this lane's edge ----
    const float x2 = x * x, y2 = y * y, z2 = z * z;
    const float xy = x * y, xz = x * z, yz = y * z;
    float m[20];
    m[0] = 1.0f; m[1] = x;       m[2] = y;       m[3] = z;
    m[4] = x2;   m[5] = y2;      m[6] = z2;      m[7] = xy;
    m[8] = xz;   m[9] = yz;      m[10] = x * x2; m[11] = y * y2;
    m[12] = z * z2; m[13] = x2 * y; m[14] = x2 * z; m[15] = x * y2;
    m[16] = y2 * z; m[17] = x * z2; m[18] = y * z2; m[19] = xy * z;

    const bool hi = lane >= 16;
    const int row = lane & 15;

    // ---- A fragments: 16x4 f32 slices of SH_C.
    // Lane l<16 holds M=l, K={4j,4j+1}; lane l>=16 holds M=l-16, K={4j+2,4j+3}.
    // Pairs are 8B-aligned in the constant table -> v2f loads.
    v2f a[5];
#pragma unroll
    for (int j = 0; j < 5; ++j) {
        const float* cp = &SH_C[row][4 * j + (hi ? 2 : 0)];
        a[j] = *(const v2f*)cp;
    }

    // ---- two 16-edge accumulators; 5 chained f32 WMMAs each (K=20) ----
    v8f acc0 = {}; // edges eBase + 0..15
    v8f acc1 = {}; // edges eBase + 16..31
#pragma unroll
    for (int j = 0; j < 5; ++j) {
        // B is 4x16: lane half fixes the K pair this lane supplies.
        const float s0 = hi ? m[4 * j + 2] : m[4 * j + 0]; // own edge, my columns
        const float s1 = hi ? m[4 * j + 3] : m[4 * j + 1];
        const float o0 = hi ? m[4 * j + 0] : m[4 * j + 2]; // own edge, partner's columns
        const float o1 = hi ? m[4 * j + 1] : m[4 * j + 3];
        const float r0 = __shfl_xor(o0, 16, 32);           // partner edge, my columns
        const float r1 = __shfl_xor(o1, 16, 32);

        v2f b0, b1;
        b0.x = hi ? r0 : s0;  b0.y = hi ? r1 : s1;  // edges 0..15 of the tile
        b1.x = hi ? s0 : r0;  b1.y = hi ? s1 : r1;  // edges 16..31 of the tile

        acc0 = __builtin_amdgcn_wmma_f32_16x16x4_f32(
                   false, a[j], false, b0, (short)0, acc0, false, false);
        acc1 = __builtin_amdgcn_wmma_f32_16x16x4_f32(
                   false, a[j], false, b1, (short)0, acc1, false, false);
    }

    // ---- store: D layout => lane holds 8 consecutive SH of one edge.
    // lanes 0-15: SH[0..7] of edge (tile,set,row); lanes 16-31: SH[8..15].
    const int coff = hi ? 8 : 0;
    const long long e0 = eBase + row;       // set 0
    const long long e1 = e0 + 16;           // set 1

    if (e0 < E) {
        float* o = out + e0 * 16 + coff;    // 16B aligned (64*e0 + {0,32})
        v4f lo = { acc0[0], acc0[1], acc0[2], acc0[3] };
        v4f hi4 = { acc0[4], acc0[5], acc0[6], acc0[7] };
        __builtin_nontemporal_store(lo, (v4f*)o);       // NT: 512MB stream, keep L2 for pos
        __builtin_nontemporal_store(hi4, (v4f*)(o + 4));
    }
    if (e1 < E) {
        float* o = out + e1 * 16 + coff;
        v4f lo = { acc1[0], acc1[1], acc1[2], acc1[3] };
        v4f hi4 = { acc1[4], acc1[5], acc1[6], acc1[7] };
        __builtin_nontemporal_store(lo, (v4f*)o);
        __builtin_nontemporal_store(hi4, (v4f*)(o + 4));
    }
}

extern "C" void kernel_launch(void* const* d_in, const int* in_sizes, int n_in,
                              void* d_out, int out_size, void* d_ws, size_t ws_size,
                              hipStream_t stream) {
    (void)n_in; (void)d_ws; (void)ws_size; (void)out_size;
    const float* pos = (const float*)d_in[0];
    const int* edge_index = (const int*)d_in[1];   // harness ABI: integer -> const int*
    float* out = (float*)d_out;

    const int E = in_sizes[1] / 2;                 // edge_index is (2, E)
    const long long tiles = ((long long)E + 31) / 32;   // 32 edges per wave
    const int blocks = (int)((tiles + 7) / 8);          // 8 waves (256 thr) per block

    sh_l3_edge_kernel<<<blocks, 256, 0, stream>>>(pos, edge_index, out, E);
}